// MultiStepTDLoss_69595650065025
// MI455X (gfx1250) — compile-verified
//
#include <hip/hip_runtime.h>

// ---------------------------------------------------------------------------
// GAE + squared TD loss, [T+1, B] time-major, fp32, B = 4096, T = 1024.
// adv[t] = A[t]*adv[t+1] + B[t]  -> chunked parallel linear-recurrence scan:
//   pass1: per-chunk local scan (carry 0) + coefficient product -> summaries
//   pass2: compose L2-resident summaries for exact carry, redo chunk, emit
//          loss = (adv + target_value - value)^2, append zero row.
// Memory-bound (~100 MB HBM, AI ~0.33 FLOP/B): WMMA does not apply; the
// win is wave32 occupancy + B128 streaming + 192 MB L2 cross-pass reuse.
// ---------------------------------------------------------------------------

#define GAMMA_F  0.99f
#define LAMBDA_F 0.95f

namespace {
constexpr int kB      = 4096;        // batch dim (reference harness)
constexpr int kBG     = kB / 4;      // float4 column groups = 1024
constexpr int kBGLog2 = 10;          // log2(kBG)
constexpr int kChunk  = 16;          // timesteps per chunk (T=1024 -> 64)
constexpr int kTPB    = 256;         // 8 wave32 per block
}

__device__ __forceinline__ float4 make4(float v) { return make_float4(v, v, v, v); }

// Per-component recurrence coefficients:
//   d1    = discount[t+1]*gamma
//   delta = reward[t+1] + d1*tv[t+1] - tv[t]
//   A     = not_last[t] * d1 * lambda ;  Bc = not_last[t] * delta
#define GAE_COEF(c)                                                            \
  {                                                                            \
    float d1    = dcv.c * GAMMA_F;                                             \
    float delta = fmaf(d1, tvn.c, rwv.c) - tv0.c;                              \
    float nl    = (stv.c == 2) ? 0.0f : 1.0f;                                  \
    A.c  = nl * (d1 * LAMBDA_F);                                               \
    Bc.c = nl * delta;                                                         \
  }

// ---------------------------------------------------------------------------
// Pass 1: per (chunk, 4-column group) compute chunk summary, stored
// interleaved: summ[2*g] = advhead (local adv at chunk head, carry-in 0),
//              summ[2*g+1] = pchunk (product of A over the chunk).
// ---------------------------------------------------------------------------
__global__ __launch_bounds__(kTPB) void gae_chunk_summary(
    const float* __restrict__ tv, const float* __restrict__ rw,
    const int* __restrict__ st, const float* __restrict__ dc,
    float4* __restrict__ summ, int T, int NC)
{
  const int g  = blockIdx.x * kTPB + threadIdx.x;
  const int j  = g & (kBG - 1);     // column group (4 columns)
  const int nc = g >> kBGLog2;      // chunk index
  if (nc >= NC) return;

  const int t0 = nc * kChunk;
  const int t1 = min(t0 + kChunk, T);

  const float4* tv4 = (const float4*)tv;
  const float4* rw4 = (const float4*)rw;
  const float4* dc4 = (const float4*)dc;
  const int4*   st4 = (const int4*)st;

  float4 adv = make4(0.0f);
  float4 P   = make4(1.0f);
  float4 tv_next = tv4[(size_t)t1 * kBG + j];  // row t1 <= T valid (T+1 rows)

#pragma unroll 4
  for (int t = t1 - 1; t >= t0; --t) {
    const size_t r  = (size_t)t * kBG + j;
    const size_t r1 = r + kBG;
    // branchless prefetch 4 rows ahead, clamped inside chunk (valid addr)
    {
      const int    tp = (t - 4 >= t0) ? (t - 4) : t0;   // gfx1250 prefetch
      const size_t rp = (size_t)tp * kBG + j;
      __builtin_prefetch(tv4 + rp,        0, 3);
      __builtin_prefetch(rw4 + rp + kBG,  0, 3);
      __builtin_prefetch(dc4 + rp + kBG,  0, 3);
      __builtin_prefetch(st4 + rp,        0, 3);
    }
    const float4 tv0 = tv4[r];
    const float4 rwv = rw4[r1];
    const float4 dcv = dc4[r1];
    const int4   stv = st4[r];
    const float4 tvn = tv_next;
    float4 A, Bc;
    GAE_COEF(x) GAE_COEF(y) GAE_COEF(z) GAE_COEF(w)
    adv.x = fmaf(A.x, adv.x, Bc.x);
    adv.y = fmaf(A.y, adv.y, Bc.y);
    adv.z = fmaf(A.z, adv.z, Bc.z);
    adv.w = fmaf(A.w, adv.w, Bc.w);
    P.x *= A.x; P.y *= A.y; P.z *= A.z; P.w *= A.w;
    tv_next = tv0;
  }
  summ[2 * (size_t)g]     = adv;   // advhead
  summ[2 * (size_t)g + 1] = P;     // pchunk  (adjacent -> one B128-pair read)
}

// ---------------------------------------------------------------------------
// Pass 2: compose later-chunk summaries (2 MB, L2-resident) to get the exact
// carry into this chunk, re-run the recurrence seeded with the true carry
// (linearity => exact values), and write loss rows. Chunk 0 also writes the
// trailing zero row T.
// ---------------------------------------------------------------------------
__global__ __launch_bounds__(kTPB) void gae_loss(
    const float* __restrict__ value, const float* __restrict__ tv,
    const float* __restrict__ rw, const int* __restrict__ st,
    const float* __restrict__ dc,
    const float4* __restrict__ summ,
    float* __restrict__ out, int T, int NC)
{
  const int g  = blockIdx.x * kTPB + threadIdx.x;
  const int j  = g & (kBG - 1);
  const int nc = g >> kBGLog2;
  if (nc >= NC) return;

  // carry into chunk nc = f_{nc+1}(f_{nc+2}(... f_{NC-1}(0)))
  // where f_k(x) = advhead[k] + pchunk[k]*x
  float4 carry = make4(0.0f);
  for (int k = NC - 1; k > nc; --k) {
    const size_t s  = 2 * ((size_t)k * kBG + j);
    const float4 ah = summ[s];
    const float4 pk = summ[s + 1];
    carry.x = fmaf(pk.x, carry.x, ah.x);
    carry.y = fmaf(pk.y, carry.y, ah.y);
    carry.z = fmaf(pk.z, carry.z, ah.z);
    carry.w = fmaf(pk.w, carry.w, ah.w);
  }

  const int t0 = nc * kChunk;
  const int t1 = min(t0 + kChunk, T);

  const float4* tv4  = (const float4*)tv;
  const float4* rw4  = (const float4*)rw;
  const float4* dc4  = (const float4*)dc;
  const float4* va4  = (const float4*)value;
  const int4*   st4  = (const int4*)st;
  float4*       out4 = (float4*)out;

  float4 adv = carry;                          // exact seed -> exact values
  float4 tv_next = tv4[(size_t)t1 * kBG + j];

#pragma unroll 4
  for (int t = t1 - 1; t >= t0; --t) {
    const size_t r  = (size_t)t * kBG + j;
    const size_t r1 = r + kBG;
    {
      const int    tp = (t - 4 >= t0) ? (t - 4) : t0;
      const size_t rp = (size_t)tp * kBG + j;
      __builtin_prefetch(tv4 + rp,        0, 3);
      __builtin_prefetch(rw4 + rp + kBG,  0, 3);
      __builtin_prefetch(dc4 + rp + kBG,  0, 3);
      __builtin_prefetch(st4 + rp,        0, 3);
      __builtin_prefetch(va4 + rp,        0, 3);
    }
    const float4 tv0 = tv4[r];
    const float4 rwv = rw4[r1];
    const float4 dcv = dc4[r1];
    const int4   stv = st4[r];
    const float4 val = va4[r];
    const float4 tvn = tv_next;
    float4 A, Bc;
    GAE_COEF(x) GAE_COEF(y) GAE_COEF(z) GAE_COEF(w)
    adv.x = fmaf(A.x, adv.x, Bc.x);
    adv.y = fmaf(A.y, adv.y, Bc.y);
    adv.z = fmaf(A.z, adv.z, Bc.z);
    adv.w = fmaf(A.w, adv.w, Bc.w);
    // loss = (adv + tv - value)^2
    float4 e;
    e.x = (adv.x + tv0.x) - val.x;
    e.y = (adv.y + tv0.y) - val.y;
    e.z = (adv.z + tv0.z) - val.z;
    e.w = (adv.w + tv0.w) - val.w;
    e.x *= e.x; e.y *= e.y; e.z *= e.z; e.w *= e.w;
    out4[r] = e;
    tv_next = tv0;
  }

  if (nc == 0) out4[(size_t)T * kBG + j] = make4(0.0f);  // appended zero row
}

extern "C" void kernel_launch(void* const* d_in, const int* in_sizes, int n_in,
                              void* d_out, int out_size, void* d_ws, size_t ws_size,
                              hipStream_t stream) {
  // setup_inputs order: value, target_value, reward, step_type, discount
  const float* value        = (const float*)d_in[0];
  const float* target_value = (const float*)d_in[1];
  const float* reward       = (const float*)d_in[2];
  const int*   step_type    = (const int*)d_in[3];
  const float* discount     = (const float*)d_in[4];
  float* out = (float*)d_out;

  const int T1 = in_sizes[0] / kB;     // 1025
  const int T  = T1 - 1;               // 1024
  const int NC = (T + kChunk - 1) / kChunk;

  float4* summ = (float4*)d_ws;        // 2 * NC * kBG float4 = 2 MB

  const int total  = NC * kBG;         // 65536 threads = 2048 wave32
  const int blocks = (total + kTPB - 1) / kTPB;

  gae_chunk_summary<<<blocks, kTPB, 0, stream>>>(
      target_value, reward, step_type, discount, summ, T, NC);
  gae_loss<<<blocks, kTPB, 0, stream>>>(
      value, target_value, reward, step_type, discount, summ, out, T, NC);
}